// SimpleFA_57501022159267
// MI455X (gfx1250) — compile-verified
//
#include <hip/hip_runtime.h>
#include <stdint.h>

// out[b,c,h,w] = alpha[slot[b,h,w],c] * x[b,c,h,w] + beta[slot[b,h,w],c]
// x: [32,2048,32,32] f32; slot: [32,32,32] i32; alpha/beta: [256,2048] f32.
#define NB     32
#define CCH    2048
#define HW     1024
#define NSLOT  256
#define BLK    256        // threads per block (8 waves)
#define CB     128        // channels per block
#define CT     32         // channels per LDS tile
#define ROWSTR (CT + 1)   // 33 floats per slot row (pad -> conflict-free banks)

typedef float    f32x4 __attribute__((ext_vector_type(4)));
typedef int      i32x4 __attribute__((ext_vector_type(4)));
typedef int      i32x8 __attribute__((ext_vector_type(8)));
typedef unsigned u32x4 __attribute__((ext_vector_type(4)));

#if __has_builtin(__builtin_amdgcn_tensor_load_to_lds)
#define HAVE_TDM 1
#endif

#if HAVE_TDM
// One TDM op: load a 32(rows) x 32(floats) tile of `gaddr` (row stride 2048 f32)
// into LDS at lds_addr, padding 1 DWORD after every 32 DWORDs -> 33-float rows.
__device__ __forceinline__ void tdm_load_tile(uint32_t lds_addr, uint64_t gaddr) {
    u32x4 g0;
    g0.x = 1u;                                               // count=1 (valid), no gather
    g0.y = lds_addr;                                         // LDS byte address
    g0.z = (uint32_t)gaddr;                                  // global_addr[31:0]
    g0.w = ((uint32_t)(gaddr >> 32) & 0x01FFFFFFu)           // global_addr[56:32]
         | (2u << 30);                                       // type=2 ("image")
    i32x8 g1;
    g1[0] = (int)((2u << 16)      // data_size = 4B
                | (1u << 20)      // pad_enable
                | (4u << 22));    // pad_interval: 32 DWORDs  (pad_amount=0 -> 1 DWORD)
    g1[1] = (int)((uint32_t)CCH << 16);      // tensor_dim0 = 2048 (bits 79:48, low half)
    g1[2] = (int)((uint32_t)NSLOT << 16);    // tensor_dim1 = 256
    g1[3] = (int)((uint32_t)CT << 16);       // tile_dim0  = 32 elements
    g1[4] = 32;                              // tile_dim1  = 32 rows, tile_dim2 = 0
    g1[5] = CCH;                             // tensor_dim0_stride = 2048
    g1[6] = 0;
    g1[7] = 0;
    i32x4 z4 = {0, 0, 0, 0};
#if __has_include(<hip/amd_detail/amd_gfx1250_TDM.h>)
    i32x8 z8 = {0, 0, 0, 0, 0, 0, 0, 0};     // therock 6-arg form
    __builtin_amdgcn_tensor_load_to_lds(g0, g1, z4, z4, z8, 0);
#else
    __builtin_amdgcn_tensor_load_to_lds(g0, g1, z4, z4, 0); // ROCm 7.2 5-arg form
#endif
}
#endif

__global__ __launch_bounds__(BLK, 4)
void slotfilm_kernel(const float* __restrict__ x,
                     const int*   __restrict__ slot_assign,
                     const float* __restrict__ alpha,
                     const float* __restrict__ beta,
                     float*       __restrict__ out)
{
    __shared__ float lds_a[NSLOT * ROWSTR];   // 33,792 B
    __shared__ float lds_b[NSLOT * ROWSTR];   // 33,792 B

    const int tid = threadIdx.x;
    const int blk = blockIdx.x;
    const int b   = blk >> 4;                 // image (32)
    const int c0  = (blk & 15) * CB;          // channel block base (16 x 128)

    // 4 consecutive pixels per thread -> b128 x/out traffic
    const i32x4 sl = *(const i32x4*)&slot_assign[b * HW + 4 * tid];

#if HAVE_TDM
    // wave id (uniform): each of the 8 waves DMAs 32 slot rows per table
    const int wid   = __builtin_amdgcn_readfirstlane(tid >> 5);
    const int srow0 = wid * 32;
#else
    const unsigned lane_c   = (unsigned)(tid & 31);
    const unsigned row_base = (unsigned)(tid >> 5);
#endif

    for (int t = 0; t < CB / CT; ++t) {
        const int ct0 = c0 + t * CT;

        // ---------------- stage alpha/beta[:, ct0..ct0+31] into LDS ----------------
#if HAVE_TDM
        tdm_load_tile((uint32_t)(uintptr_t)&lds_a[srow0 * ROWSTR],
                      (uint64_t)(uintptr_t)(alpha + (size_t)srow0 * CCH + ct0));
        tdm_load_tile((uint32_t)(uintptr_t)&lds_b[srow0 * ROWSTR],
                      (uint64_t)(uintptr_t)(beta  + (size_t)srow0 * CCH + ct0));
#if __has_builtin(__builtin_amdgcn_s_wait_tensorcnt)
        __builtin_amdgcn_s_wait_tensorcnt(0);
#else
        asm volatile("s_wait_tensorcnt 0x0" ::: "memory");
#endif
#else   // fallback: per-lane async global->LDS b32
#pragma unroll 4
        for (int k = 0; k < 32; ++k) {
            const unsigned srow  = row_base + 8u * (unsigned)k;   // 0..255
            const unsigned loff  = (srow * ROWSTR + lane_c) * 4u;
            const unsigned lds_A = (unsigned)(uintptr_t)lds_a + loff;
            const unsigned lds_B = (unsigned)(uintptr_t)lds_b + loff;
            const uint64_t ga = (uint64_t)(uintptr_t)(alpha + (size_t)srow * CCH + ct0 + lane_c);
            const uint64_t gb = (uint64_t)(uintptr_t)(beta  + (size_t)srow * CCH + ct0 + lane_c);
            asm volatile("global_load_async_to_lds_b32 %0, %1, off" :: "v"(lds_A), "v"(ga) : "memory");
            asm volatile("global_load_async_to_lds_b32 %0, %1, off" :: "v"(lds_B), "v"(gb) : "memory");
        }
#if __has_builtin(__builtin_amdgcn_s_wait_asynccnt)
        __builtin_amdgcn_s_wait_asynccnt(0);
#else
        asm volatile("s_wait_asynccnt 0x0" ::: "memory");
#endif
#endif
        __syncthreads();   // all waves' DMA chunks visible

        // ---------------- compute: b128 streaming + LDS gathers ----------------
        const int r0 = sl.x * ROWSTR, r1 = sl.y * ROWSTR,
                  r2 = sl.z * ROWSTR, r3 = sl.w * ROWSTR;
#pragma unroll 4
        for (int cc = 0; cc < CT; ++cc) {
            const size_t xi = ((size_t)b * CCH + (ct0 + cc)) * HW + 4u * (unsigned)tid;
            const f32x4 xv = __builtin_nontemporal_load((const f32x4*)(x + xi));
            f32x4 r;
            r.x = fmaf(lds_a[r0 + cc], xv.x, lds_b[r0 + cc]);
            r.y = fmaf(lds_a[r1 + cc], xv.y, lds_b[r1 + cc]);
            r.z = fmaf(lds_a[r2 + cc], xv.z, lds_b[r2 + cc]);
            r.w = fmaf(lds_a[r3 + cc], xv.w, lds_b[r3 + cc]);
            __builtin_nontemporal_store(r, (f32x4*)(out + xi));
        }
        __syncthreads();   // tile consumed before next DMA overwrites LDS
    }
}

extern "C" void kernel_launch(void* const* d_in, const int* in_sizes, int n_in,
                              void* d_out, int out_size, void* d_ws, size_t ws_size,
                              hipStream_t stream) {
    const float* x           = (const float*)d_in[0];
    const int*   slot_assign = (const int*)  d_in[1];
    const float* alpha       = (const float*)d_in[2];
    const float* beta        = (const float*)d_in[3];
    float*       out         = (float*)d_out;

    dim3 grid(NB * (CCH / CB));   // 512 blocks
    dim3 block(BLK);              // 256 threads = 8 wave32
    slotfilm_kernel<<<grid, block, 0, stream>>>(x, slot_assign, alpha, beta, out);
}